// SparseFcaf3DNeckWithHead_87153476370686
// MI455X (gfx1250) — compile-verified
//
#include <hip/hip_runtime.h>
#include <hip/hip_bf16.h>

typedef float v2f __attribute__((ext_vector_type(2)));
typedef float v8f __attribute__((ext_vector_type(8)));

#define MAXB 16

// Monotone float -> uint key (order-preserving for all finite floats)
__device__ __forceinline__ unsigned fkey(float s) {
    unsigned u = __float_as_uint(s);
    return (u & 0x80000000u) ? ~u : (u | 0x80000000u);
}

// ---------------------------------------------------------------------------
// Kernel 1: per-batch k-th-largest score key via 4-round radix-256 select.
// Single workgroup; scores (1MB) are L2-resident so 4 passes are cheap.
// Matches reference semantics: batches with < k members get kth = key(-1e30).
// ---------------------------------------------------------------------------
__global__ __launch_bounds__(1024)
void topk_select_kernel(const float* __restrict__ scores,
                        const int* __restrict__ batch_ids,
                        const int* __restrict__ nb_ptr,
                        const int* __restrict__ k_ptr,
                        int N, unsigned* __restrict__ kth_out) {
    __shared__ int hist[MAXB][256];
    __shared__ int cnt[MAXB];
    __shared__ unsigned prefix[MAXB];
    __shared__ int remk[MAXB];

    const int tid = threadIdx.x;
    const int T = blockDim.x;
    const int B = *nb_ptr;
    const int K = *k_ptr;

    if (tid < MAXB) { cnt[tid] = 0; prefix[tid] = 0u; remk[tid] = K; }
    __syncthreads();

    // Per-batch member counts. batch_ids is sorted, so the strided subsequence
    // seen by each thread is also sorted -> at most B shared atomics/thread.
    {
        int curb = -1, c = 0;
        for (int i = tid; i < N; i += T) {
            int b = batch_ids[i];
            if (b != curb) { if (c) atomicAdd(&cnt[curb], c); curb = b; c = 0; }
            c++;
        }
        if (c) atomicAdd(&cnt[curb], c);
    }
    __syncthreads();

    for (int r = 0; r < 4; ++r) {
        const int shift = 24 - 8 * r;
        for (int i = tid; i < MAXB * 256; i += T) ((int*)hist)[i] = 0;
        __syncthreads();

        for (int i = tid; i < N; i += T) {
            int b = batch_ids[i];
            if (cnt[b] < K) continue;
            unsigned key = fkey(scores[i]);
            bool match;
            if (r == 0) {
                match = true;
            } else {
                int hs = shift + 8;
                match = (key >> hs) == (prefix[b] >> hs);
            }
            if (match) atomicAdd(&hist[b][(key >> shift) & 255], 1);
        }
        __syncthreads();

        if (tid < B && cnt[tid] >= K) {
            int acc = 0;
            int rk = remk[tid];
            for (int bin = 255; bin >= 0; --bin) {
                int c = hist[tid][bin];
                if (acc + c >= rk) {
                    prefix[tid] |= ((unsigned)bin) << shift;
                    remk[tid] = rk - acc;
                    break;
                }
                acc += c;
            }
        }
        __syncthreads();
    }

    if (tid < B) {
        kth_out[tid] = (cnt[tid] >= K) ? prefix[tid] : fkey(-1e30f);
    }
}

// ---------------------------------------------------------------------------
// Kernel 2: masked [N,256]x[256,24] GEMM with exp/bias epilogue.
// 8 waves/block, 16 rows per wave (128 rows/block), V_WMMA_F32_16X16X4_F32.
// Weights staged in LDS in K-pair-interleaved layout so every B fragment is
// one aligned ds_load_b64. K-loop unrolled 4x for memory-level parallelism.
// ---------------------------------------------------------------------------
__global__ __launch_bounds__(256)
void head_gemm_kernel(const float* __restrict__ feats,
                      const float* __restrict__ scores,
                      const int* __restrict__ batch_ids,
                      const float* __restrict__ w_reg,
                      const float* __restrict__ w_cls,
                      const float* __restrict__ b_cls,
                      const float* __restrict__ scale,
                      const unsigned* __restrict__ kth,
                      float* __restrict__ out,
                      int N, int C) {
    // Wp[kp][col][j] = W[2*kp + j][col]; W cols 0..5 = w_reg, 6..23 = w_cls, rest 0
    __shared__ __align__(16) float Wp[128][32][2];
    __shared__ float maskv[128];
    __shared__ float shb[32];      // padded bias per output column (0 for reg cols)
    __shared__ float shsc;

    const int tid = threadIdx.x;
    const int row0 = blockIdx.x * 128;
    const int Cc = (C > 256) ? 256 : C;

    for (int idx = tid; idx < 256 * 32; idx += 256) {
        int rr = idx >> 5, cc = idx & 31;    // rr = K index, cc = output column
        float v = 0.f;
        if (rr < Cc) {
            if (cc < 6)       v = w_reg[rr * 6 + cc];
            else if (cc < 24) v = w_cls[rr * 18 + (cc - 6)];
        }
        Wp[rr >> 1][cc][rr & 1] = v;
    }
    if (tid < 128) {
        int r = row0 + tid;
        float m = 0.f;
        if (r < N) {
            unsigned key = fkey(scores[r]);
            m = (key >= kth[batch_ids[r]]) ? 1.f : 0.f;
        }
        maskv[tid] = m;
    }
    if (tid < 32) shb[tid] = (tid >= 6 && tid < 24) ? b_cls[tid - 6] : 0.f;
    if (tid == 0) shsc = scale[0];
    __syncthreads();

    const int wave = tid >> 5;
    const int lane = tid & 31;
    const int col0 = lane & 15;
    const int koff = (lane >> 4) << 1;        // lanes 0-15: K0/K1, lanes 16-31: K2/K3
    const int rbase = row0 + wave * 16;

    int arow = rbase + (lane & 15);
    if (arow >= N) arow = N - 1;              // keep loads in-bounds; stores guarded below
    const float* ap = feats + (size_t)arow * (size_t)C + koff;

    v8f acc0 = {};
    v8f acc1 = {};

#pragma unroll 4
    for (int k = 0; k < Cc; k += 4) {
        v2f a = *(const v2f*)(ap + k);                     // 8B coalesced A fragment
        int kp = (k + koff) >> 1;
        v2f b0 = *(const v2f*)(&Wp[kp][col0][0]);          // ds_load_b64
        v2f b1 = *(const v2f*)(&Wp[kp][col0 + 16][0]);     // ds_load_b64
        acc0 = __builtin_amdgcn_wmma_f32_16x16x4_f32(
            false, a, false, b0, (short)0, acc0, false, false);
        acc1 = __builtin_amdgcn_wmma_f32_16x16x4_f32(
            false, a, false, b1, (short)0, acc1, false, false);
    }

    // Epilogue: C/D layout -> VGPR v holds row rbase+v (lanes 0-15) / rbase+8+v (16-31)
    const float sc = shsc;
    const int rhalf = rbase + ((lane >> 4) << 3);
    const bool isreg0 = (col0 < 6);
    const float bias0 = shb[col0];
    const int col1 = col0 + 16;
    const bool valid1 = (col1 < 24);
    const float bias1 = shb[col1];

    for (int v = 0; v < 8; ++v) {
        int r = rhalf + v;
        if (r >= N) break;
        float m = maskv[r - row0];
        float t0 = acc0[v];
        float o0 = isreg0 ? __expf(sc * m * t0) : (m * t0 + bias0);
        out[(size_t)r * 24 + col0] = o0;
        if (valid1) {
            float t1 = acc1[v];
            out[(size_t)r * 24 + col1] = m * t1 + bias1;
        }
    }
}

extern "C" void kernel_launch(void* const* d_in, const int* in_sizes, int n_in,
                              void* d_out, int out_size, void* d_ws, size_t ws_size,
                              hipStream_t stream) {
    const float* feats     = (const float*)d_in[0];
    const float* scores    = (const float*)d_in[1];
    const float* w_reg     = (const float*)d_in[2];
    const float* w_cls     = (const float*)d_in[3];
    const float* b_cls     = (const float*)d_in[4];
    const float* scale     = (const float*)d_in[5];
    const int*   batch_ids = (const int*)d_in[6];
    const int*   nb_ptr    = (const int*)d_in[7];
    const int*   k_ptr     = (const int*)d_in[8];

    const int N = in_sizes[1];            // scores is [N,1]
    const int C = in_sizes[2] / 6;        // w_reg is [C,6]

    unsigned* kth_keys = (unsigned*)d_ws; // 64 bytes of scratch

    topk_select_kernel<<<1, 1024, 0, stream>>>(scores, batch_ids, nb_ptr, k_ptr,
                                               N, kth_keys);

    const int blocks = (N + 127) / 128;
    head_gemm_kernel<<<blocks, 256, 0, stream>>>(feats, scores, batch_ids,
                                                 w_reg, w_cls, b_cls, scale,
                                                 kth_keys, (float*)d_out, N, C);
}